// AladynSurvivalFixedKernelsAvgLoss_5514738008970
// MI455X (gfx1250) — compile-verified
//
#include <hip/hip_runtime.h>
#include <math.h>

// ---------------------------------------------------------------------------
// AladynSurvival fixed-kernels average loss for MI455X (gfx1250, wave32)
//
// N=4000, D=256, T=52, K=20, P=100.  Strategy:
//   * softmax(lambda) -> thetaT [t][n][k]      (16.6 MB ws)
//   * sigmoid(phi)    -> phiT   [t][k][d]      (1.06 MB ws)
//   * fused GEMM+loss: one wave per 16x16 (n,d) tile per t,
//     5x V_WMMA_F32_16X16X4_F32 over K=20, then masked-log loss in-register
//     (pi is never materialized; Y is only gathered at t==event_time)
//   * GP prior via trace identity: 0.5 * sum(Kinv o Gram(dev)),
//     Gram(52x52) accumulated in LDS-tiled blocks, Kinv built on device
//     (power iteration for cond estimate -> jitter loop -> Gauss-Jordan)
// ---------------------------------------------------------------------------

#define Nn 4000
#define Dd 256
#define Tt 52
#define Kk 20
#define Pp 100
#define NKROWS (Nn * Kk)   // 80000
#define KDROWS (Kk * Dd)   // 5120
#define EPSF 1e-8f

typedef float v2f __attribute__((ext_vector_type(2)));
typedef float v8f __attribute__((ext_vector_type(8)));

// ---------------- workspace layout (bytes, 256-aligned) ----------------
#define OFF_THETA 0ul                     // N*K*T floats = 16,640,000
#define OFF_PHIT  16640000ul              // K*D*T floats =  1,064,960
#define OFF_MEAN  17704960ul              // N*K   floats =    320,000
#define OFF_MLAM  18024960ul              // 52*52 floats (padded 11008)
#define OFF_MPHI  18035968ul
#define OFF_ALAM  18046976ul
#define OFF_APHI  18057984ul
#define OFF_SUM   18068992ul              // 1 float

// ---------------------------------------------------------------------------
__global__ void zero_kernel(float* Mlam, float* Mphi, float* dsum) {
    int i = blockIdx.x * blockDim.x + threadIdx.x;
    if (i < Tt * Tt) { Mlam[i] = 0.0f; Mphi[i] = 0.0f; }
    if (i == 0) dsum[0] = 0.0f;
}

// ---------------------------------------------------------------------------
// theta = softmax over K; write transposed layout thetaT[(t*N + n)*K + k]
__global__ void theta_kernel(const float* __restrict__ lam,
                             float* __restrict__ thetaT) {
    int idx = blockIdx.x * blockDim.x + threadIdx.x;   // idx = t*N + n
    if (idx >= Tt * Nn) return;
    int t = idx / Nn, n = idx % Nn;
    float v[Kk];
    float m = -1e30f;
    #pragma unroll
    for (int k = 0; k < Kk; ++k) {
        v[k] = lam[((size_t)n * Kk + k) * Tt + t];
        m = fmaxf(m, v[k]);
    }
    float s = 0.0f;
    #pragma unroll
    for (int k = 0; k < Kk; ++k) { v[k] = expf(v[k] - m); s += v[k]; }
    float inv = 1.0f / s;
    float* o = thetaT + (size_t)idx * Kk;
    #pragma unroll
    for (int k = 0; k < Kk; ++k) o[k] = v[k] * inv;
}

// sigmoid(phi) transposed: phiT[(t*K + k)*D + d]
__global__ void phiT_kernel(const float* __restrict__ phi,
                            float* __restrict__ phiT) {
    int idx = blockIdx.x * blockDim.x + threadIdx.x;
    if (idx >= Tt * Kk * Dd) return;
    int d = idx % Dd;
    int k = (idx / Dd) % Kk;
    int t = idx / (Dd * Kk);
    float x = phi[((size_t)k * Dd + d) * Tt + t];
    phiT[idx] = 1.0f / (1.0f + expf(-x));
}

// mean_lam[n*K + k] = sum_p G[n,p] * gamma[p,k]
__global__ void mean_lam_kernel(const float* __restrict__ G,
                                const float* __restrict__ gamma,
                                float* __restrict__ mean) {
    int idx = blockIdx.x * blockDim.x + threadIdx.x;
    if (idx >= NKROWS) return;
    int n = idx / Kk, k = idx % Kk;
    float s = 0.0f;
    for (int p = 0; p < Pp; ++p) s += G[n * Pp + p] * gamma[p * Kk + k];
    mean[idx] = s;
}

// ---------------------------------------------------------------------------
// Fused pi-GEMM + survival loss.  One wave32 per 16x16 (n,d) tile per t.
// A = theta_t (16x20 f32), B = phi_prob_t (20x16 f32), 5x wmma_f32_16x16x4_f32.
__global__ __launch_bounds__(256)
void loss_kernel(const float* __restrict__ thetaT,
                 const float* __restrict__ phiT,
                 const float* __restrict__ Y,
                 const int*   __restrict__ ev,
                 float* __restrict__ sumOut) {
    __shared__ float waveSum[8];
    const int wave = threadIdx.x >> 5;
    const int lane = threadIdx.x & 31;
    const int lo = lane & 15, hi = lane >> 4;

    const int tile = blockIdx.x * 8 + wave;            // 0 .. 207999
    const int t   = tile / (Nn / 16 * Dd / 16);        // / 4000
    const int rem = tile % (Nn / 16 * Dd / 16);
    const int n0 = (rem >> 4) << 4;                    // 250 n-tiles
    const int d0 = (rem & 15) << 4;                    // 16 d-tiles

    const float* thT = thetaT + (size_t)t * (Nn * Kk); // [n][k], stride K
    const float* phT = phiT   + (size_t)t * (Kk * Dd); // [k][d], stride D

    v8f acc = {0.f, 0.f, 0.f, 0.f, 0.f, 0.f, 0.f, 0.f};
    #pragma unroll
    for (int k0 = 0; k0 < Kk; k0 += 4) {
        // A 16x4 f32: lanes 0-15 -> K=k0,k0+1 ; lanes 16-31 -> K=k0+2,k0+3
        const float* ap = thT + (size_t)(n0 + lo) * Kk + k0 + 2 * hi;
        v2f a; a.x = ap[0]; a.y = ap[1];
        // B 4x16 f32: lane-half selects K pair, column = d0+lo
        const float* bp = phT + (size_t)(k0 + 2 * hi) * Dd + d0 + lo;
        v2f b; b.x = bp[0]; b.y = bp[Dd];
        acc = __builtin_amdgcn_wmma_f32_16x16x4_f32(
            false, a, false, b, (short)0, acc, false, false);
    }

    // C/D layout: value r -> (n = n0 + r + 8*hi, d = d0 + lo)
    float s = 0.0f;
    #pragma unroll
    for (int r = 0; r < 8; ++r) {
        const int n = n0 + r + 8 * hi;
        const int d = d0 + lo;
        float pi = acc[r];
        pi = fminf(fmaxf(pi, EPSF), 1.0f - EPSF);
        const int e = ev[n * Dd + d];
        if (t < e) {
            s -= log1pf(-pi);                           // censored interval
        } else if (t == e) {
            const float y = Y[((size_t)n * Dd + d) * Tt + t];
            s -= y * logf(pi) + (1.0f - y) * log1pf(-pi);
        }
    }
    #pragma unroll
    for (int off = 16; off; off >>= 1) s += __shfl_xor(s, off, 32);
    if (lane == 0) waveSum[wave] = s;
    __syncthreads();
    if (threadIdx.x == 0) {
        float tot = 0.0f;
        #pragma unroll
        for (int w = 0; w < 8; ++w) tot += waveSum[w];
        atomicAdd(sumOut, tot);
    }
}

// ---------------------------------------------------------------------------
// 52x52 Gram matrices of deviation rows (upper triangle, atomically summed).
__global__ __launch_bounds__(256)
void gram_lambda(const float* __restrict__ lam,
                 const float* __restrict__ mean,
                 float* __restrict__ Mout) {
    __shared__ float rows[64][Tt + 1];
    const int tid = threadIdx.x;
    const int rowBase = blockIdx.x * 64;
    for (int idx = tid; idx < 64 * Tt; idx += 256) {
        int r = idx / Tt, t = idx % Tt;
        int row = rowBase + r;
        rows[r][t] = lam[(size_t)row * Tt + t] - mean[row];
    }
    __syncthreads();
    for (int p = tid; p < Tt * Tt; p += 256) {
        int t1 = p / Tt, t2 = p % Tt;
        if (t1 > t2) continue;
        float s = 0.0f;
        for (int r = 0; r < 64; ++r) s += rows[r][t1] * rows[r][t2];
        atomicAdd(&Mout[p], s);
    }
}

__global__ __launch_bounds__(256)
void gram_phi(const float* __restrict__ phi,
              const float* __restrict__ logit_prev,
              float* __restrict__ Mout) {
    __shared__ float rows[64][Tt + 1];
    const int tid = threadIdx.x;
    const int rowBase = blockIdx.x * 64;               // rows over K*D
    for (int idx = tid; idx < 64 * Tt; idx += 256) {
        int r = idx / Tt, t = idx % Tt;
        int row = rowBase + r;
        int dd = row % Dd;
        rows[r][t] = phi[(size_t)row * Tt + t] - logit_prev[dd * Tt + t];
    }
    __syncthreads();
    for (int p = tid; p < Tt * Tt; p += 256) {
        int t1 = p / Tt, t2 = p % Tt;
        if (t1 > t2) continue;
        float s = 0.0f;
        for (int r = 0; r < 64; ++r) s += rows[r][t1] * rows[r][t2];
        atomicAdd(&Mout[p], s);
    }
}

// ---------------------------------------------------------------------------
// Rebuild the fixed RBF kernels + jitter loop + 52x52 inverse, on device.
// blockIdx.x: 0 -> K_LAMBDA (ls = 13), 1 -> K_PHI (ls = 52/3).
// cond(K + jI) ~ (lam_max + j)/j since lam_min(K_f32) is O(1e-6) << j.
__global__ __launch_bounds__(64)
void prep_gp(float* __restrict__ A_lam, float* __restrict__ A_phi) {
    __shared__ double M[Tt][2 * Tt];     // augmented [K+jI | I], 43.3 KB
    __shared__ double v[Tt], w[Tt];
    __shared__ double lamMax;
    const int tid = threadIdx.x;
    const float ls = (blockIdx.x == 0) ? 13.0f : (float)(52.0 / 3.0);
    const float c = -0.5f / (ls * ls);

    // power iteration for lam_max of the float32 RBF matrix
    if (tid < Tt) v[tid] = 1.0;
    __syncthreads();
    for (int it = 0; it < 100; ++it) {
        if (tid < Tt) {
            double acc = 0.0;
            for (int j = 0; j < Tt; ++j) {
                float dd = (float)(tid - j);
                float kv = expf(c * dd * dd);          // float32 entries
                acc += (double)kv * v[j];
            }
            w[tid] = acc;
        }
        __syncthreads();
        if (tid == 0) {
            double nrm = 0.0;
            for (int j = 0; j < Tt; ++j) nrm += w[j] * w[j];
            nrm = sqrt(nrm);
            lamMax = nrm;
            for (int j = 0; j < Tt; ++j) v[j] = w[j] / nrm;
        }
        __syncthreads();
    }

    // jitter-doubling loop (mirrors the reference's cond() loop)
    double jitter = 0.0001;
    while (true) {
        double cond = (lamMax + jitter) / jitter;
        if (cond < 10000.0) break;
        jitter *= 2.0;
        if (jitter > 0.1) break;
    }
    const float jf = (float)jitter;

    // build augmented system
    for (int idx = tid; idx < Tt * Tt; idx += 64) {
        int i = idx / Tt, j = idx % Tt;
        float dd = (float)(i - j);
        float kv = expf(c * dd * dd);
        if (i == j) kv += jf;
        M[i][j] = (double)kv;
        M[i][Tt + j] = (i == j) ? 1.0 : 0.0;
    }
    __syncthreads();

    // Gauss-Jordan (SPD -> no pivoting needed)
    for (int p = 0; p < Tt; ++p) {
        double pivinv = 1.0 / M[p][p];
        __syncthreads();
        for (int cc = tid; cc < 2 * Tt; cc += 64) M[p][cc] *= pivinv;
        __syncthreads();
        if (tid < Tt && tid != p) {
            double f = M[tid][p];
            for (int cc = 0; cc < 2 * Tt; ++cc) M[tid][cc] -= f * M[p][cc];
        }
        __syncthreads();
    }

    float* out = (blockIdx.x == 0) ? A_lam : A_phi;
    for (int idx = tid; idx < Tt * Tt; idx += 64) {
        int i = idx / Tt, j = idx % Tt;
        out[idx] = (float)M[i][Tt + j];
    }
}

// ---------------------------------------------------------------------------
__global__ void final_kernel(const float* __restrict__ dsum,
                             const float* __restrict__ Alam,
                             const float* __restrict__ Mlam,
                             const float* __restrict__ Aphi,
                             const float* __restrict__ Mphi,
                             float* __restrict__ out) {
    if (threadIdx.x != 0 || blockIdx.x != 0) return;
    double gl = 0.0, gp = 0.0;
    for (int t1 = 0; t1 < Tt; ++t1)
        for (int t2 = t1; t2 < Tt; ++t2) {
            double wgt = (t1 == t2) ? 1.0 : 2.0;   // Gram stored upper-tri
            int p = t1 * Tt + t2;
            gl += wgt * (double)Alam[p] * (double)Mlam[p];
            gp += wgt * (double)Aphi[p] * (double)Mphi[p];
        }
    double dataLoss = (double)dsum[0] / (double)Nn;
    double gpLoss = 0.5 * gl / (double)Nn + 0.5 * gp / (double)Dd;
    out[0] = (float)(dataLoss + gpLoss);
}

// ---------------------------------------------------------------------------
extern "C" void kernel_launch(void* const* d_in, const int* in_sizes, int n_in,
                              void* d_out, int out_size, void* d_ws, size_t ws_size,
                              hipStream_t stream) {
    const float* lambda_    = (const float*)d_in[0];  // (N,K,T)
    const float* phi        = (const float*)d_in[1];  // (K,D,T)
    const float* gamma      = (const float*)d_in[2];  // (P,K)
    const float* G          = (const float*)d_in[3];  // (N,P)
    const float* Y          = (const float*)d_in[4];  // (N,D,T)
    const float* logit_prev = (const float*)d_in[5];  // (D,T)
    const int*   ev         = (const int*)d_in[6];    // (N,D)

    char* ws = (char*)d_ws;
    float* thetaT = (float*)(ws + OFF_THETA);
    float* phiT   = (float*)(ws + OFF_PHIT);
    float* meanL  = (float*)(ws + OFF_MEAN);
    float* Mlam   = (float*)(ws + OFF_MLAM);
    float* Mphi   = (float*)(ws + OFF_MPHI);
    float* Alam   = (float*)(ws + OFF_ALAM);
    float* Aphi   = (float*)(ws + OFF_APHI);
    float* dsum   = (float*)(ws + OFF_SUM);
    float* out    = (float*)d_out;

    zero_kernel<<<(Tt * Tt + 255) / 256, 256, 0, stream>>>(Mlam, Mphi, dsum);
    prep_gp<<<2, 64, 0, stream>>>(Alam, Aphi);
    theta_kernel<<<(Tt * Nn + 255) / 256, 256, 0, stream>>>(lambda_, thetaT);
    phiT_kernel<<<(Tt * Kk * Dd + 255) / 256, 256, 0, stream>>>(phi, phiT);
    mean_lam_kernel<<<(NKROWS + 255) / 256, 256, 0, stream>>>(G, gamma, meanL);
    gram_lambda<<<NKROWS / 64, 256, 0, stream>>>(lambda_, meanL, Mlam);
    gram_phi<<<KDROWS / 64, 256, 0, stream>>>(phi, logit_prev, Mphi);
    loss_kernel<<<(Tt * (Nn / 16) * (Dd / 16)) / 8, 256, 0, stream>>>(
        thetaT, phiT, Y, ev, dsum);
    final_kernel<<<1, 64, 0, stream>>>(dsum, Alam, Mlam, Aphi, Mphi, out);
}